// SymmetryFunction_8134668059077
// MI455X (gfx1250) — compile-verified
//
#include <hip/hip_runtime.h>
#include <math.h>

typedef float v2f __attribute__((ext_vector_type(2)));
typedef float v8f __attribute__((ext_vector_type(8)));

#define PI_F      3.14159265358979323846f
#define R_CUT_F   5.2f
#define A_CUT_F   3.5f
#define N_ELEM    4
#define N_PT      10      // N_PAIR_TYPES
#define ROW       1344    // 64 radial + 1280 angular
#define NMU_R     16

// ---------------------------------------------------------------- utilities
__device__ __forceinline__ float fcut(float d, float rc) {
    return 0.5f * __cosf(d * (PI_F / rc)) + 0.5f;
}

// fast pow for base in [0,1], positive exponent: exp2(z*log2(b)); b==0 -> 0 correctly
// (v_log_f32 is log2; v_exp_f32 is exp2)
__device__ __forceinline__ float fast_pow01(float b, float z) {
    return __builtin_amdgcn_exp2f(z * __builtin_amdgcn_logf(b));
}

__device__ __forceinline__ int lower_bound_i(const int* a, int n, int key) {
    int lo = 0, hi = n;
    while (lo < hi) { int mid = (lo + hi) >> 1; if (a[mid] < key) lo = mid + 1; else hi = mid; }
    return lo;
}

// ---------------------------------------------------------------- kernels
__global__ void zero_out_k(float* out, size_t n) {
    size_t i = (size_t)blockIdx.x * blockDim.x + threadIdx.x;
    size_t stride = (size_t)gridDim.x * blockDim.x;
    for (; i < n; i += stride) out[i] = 0.0f;
}

// per-pair: vec (pos[j]-pos[i]), dist, packed species (ei | ej<<4)
__global__ void prep_pairs_k(const float* __restrict__ pos,
                             const int* __restrict__ pidx,
                             const int* __restrict__ elem,
                             float4* __restrict__ pv,
                             int* __restrict__ es,
                             int npair) {
    int p = blockIdx.x * blockDim.x + threadIdx.x;
    if (p >= npair) return;
    int i = pidx[p];
    int j = pidx[npair + p];
    float vx = pos[3 * j + 0] - pos[3 * i + 0];
    float vy = pos[3 * j + 1] - pos[3 * i + 1];
    float vz = pos[3 * j + 2] - pos[3 * i + 2];
    float d  = sqrtf(vx * vx + vy * vy + vz * vz);
    pv[p] = make_float4(vx, vy, vz, d);
    es[p] = (elem[i] & 15) | ((elem[j] & 15) << 4);
}

// radial AEV: 16 lanes per pair, one mu each, two atomic adds
__global__ void radial_k(const float4* __restrict__ pv,
                         const int* __restrict__ es,
                         const int* __restrict__ pidx,
                         const float* __restrict__ reta,
                         const float* __restrict__ rmu,
                         float* __restrict__ out,
                         int npair) {
    int gid = blockIdx.x * blockDim.x + threadIdx.x;
    int p = gid >> 4;
    int k = gid & 15;
    if (p >= npair) return;
    float d = pv[p].w;
    float t = d - rmu[k];
    float val = __expf(-reta[0] * t * t) * fcut(d, R_CUT_F);
    int sp = es[p];
    int ei = sp & 15, ej = sp >> 4;
    int i = pidx[p], j = pidx[npair + p];
    atomicAdd(out + (size_t)i * ROW + ej * NMU_R + k, val);
    atomicAdd(out + (size_t)j * ROW + ei * NMU_R + k, val);
}

struct Tri { float cosd, sind, p2; int enc; };

__device__ __forceinline__ Tri eval_tri(int t, int tEnd, int ntri,
                                        const int* __restrict__ tpair,
                                        const int* __restrict__ tflip,
                                        const float4* __restrict__ pv,
                                        const int* __restrict__ es,
                                        float aeta_l, float amu_l) {
    Tri r;
    bool valid = (t < tEnd);
    int tt = valid ? t : (tEnd - 1);          // rounds>0 guarantees tEnd-1 valid
    int pa = tpair[tt], pb = tpair[ntri + tt];
    int f0 = tflip[tt], f1 = tflip[ntri + tt];
    float4 va = pv[pa];
    float4 vb = pv[pb];
    float s = ((f0 != 0) != (f1 != 0)) ? -1.0f : 1.0f;
    float dotv = va.x * vb.x + va.y * vb.y + va.z * vb.z;
    float cosang = __fdividef(s * dotv, fmaxf(va.w * vb.w, 1e-9f));
    float cosd = 0.99f * cosang;                         // cos(arccos(0.99*cosang))
    float sind = sqrtf(fmaxf(1.0f - cosd * cosd, 0.0f)); // sin(...), >=0
    float fc2 = fcut(va.w, A_CUT_F) * fcut(vb.w, A_CUT_F);
    float x = (va.w + vb.w) * 0.5f - amu_l;
    float p2 = __expf(-aeta_l * x * x) * fc2;
    int sa = es[pa], sb = es[pb];
    int eA = f0 ? (sa & 15) : (sa >> 4);                 // other0 = flip ? ti : tj
    int eB = f1 ? (sb & 15) : (sb >> 4);
    int mn = min(eA, eB), mx = max(eA, eB);
    r.enc  = valid ? (mn * N_ELEM - ((mn * (mn + 1)) >> 1) + mx) : -1;
    r.cosd = valid ? cosd : 0.0f;
    r.sind = valid ? sind : 0.0f;
    r.p2   = valid ? p2   : 0.0f;                        // keep B finite & zero when padded
    return r;
}

// one wave (32 lanes) per center atom; angular accumulation as one-hot WMMA GEMM:
// D[enc(16), j(128)] += onehot(16 x 4) * ang(4 x 128)  via 8 tiles of V_WMMA_F32_16X16X4_F32
__global__ __launch_bounds__(32) void angular_k(const float4* __restrict__ pv,
                                                const int* __restrict__ es,
                                                const int* __restrict__ tcenter,
                                                const int* __restrict__ tpair,
                                                const int* __restrict__ tflip,
                                                const float* __restrict__ anu,
                                                const float* __restrict__ azeta,
                                                const float* __restrict__ aeta,
                                                const float* __restrict__ amu,
                                                float* __restrict__ out,
                                                int ntri) {
    int c    = blockIdx.x;
    int lane = threadIdx.x;

    int tStart = lower_bound_i(tcenter, ntri, c);
    int tEnd   = lower_bound_i(tcenter, ntri, c + 1);

    // lane column inside a 16-wide tile: n = zeta*8 + eta*4 + mu  (tile index == nu)
    int n = lane & 15;
    float zeta_l = azeta[(n >> 3) & 1];
    float aeta_l = aeta[(n >> 2) & 1];
    float amu_l  = amu[n & 3];

    float cnu[8], snu[8];
#pragma unroll
    for (int q = 0; q < 8; ++q) { float nv = anu[q]; cnu[q] = __cosf(nv); snu[q] = __sinf(nv); }

    v8f acc[8];
#pragma unroll
    for (int T = 0; T < 8; ++T)
#pragma unroll
        for (int r2 = 0; r2 < 8; ++r2) acc[T][r2] = 0.0f;

    int cnt = tEnd - tStart;
    int rounds = (cnt + 3) >> 2;

    for (int rd = 0; rd < rounds; ++rd) {
        // lanes 0-15 carry K=0,1 ; lanes 16-31 carry K=2,3 (f32 16x16x4 layout)
        int tA = tStart + rd * 4 + ((lane < 16) ? 0 : 2);
        Tri t0 = eval_tri(tA,     tEnd, ntri, tpair, tflip, pv, es, aeta_l, amu_l);
        Tri t1 = eval_tri(tA + 1, tEnd, ntri, tpair, tflip, pv, es, aeta_l, amu_l);

        // one-hot A: row m = enc of the triple
        v2f A;
        A.x = (t0.enc == n) ? 1.0f : 0.0f;
        A.y = (t1.enc == n) ? 1.0f : 0.0f;

#pragma unroll
        for (int T = 0; T < 8; ++T) {
            float b0 = fmaxf(fmaf(t0.sind, snu[T], fmaf(t0.cosd, cnu[T], 1.0f)) * 0.5f, 0.0f);
            float b1 = fmaxf(fmaf(t1.sind, snu[T], fmaf(t1.cosd, cnu[T], 1.0f)) * 0.5f, 0.0f);
            v2f B;
            B.x = fast_pow01(b0, zeta_l) * t0.p2;
            B.y = fast_pow01(b1, zeta_l) * t1.p2;
            acc[T] = __builtin_amdgcn_wmma_f32_16x16x4_f32(
                false, A, false, B, (short)0, acc[T], false, false);
        }
    }

    // flush: D VGPR r of lane L holds M = r + (L<16?0:8), N = L&15; rows >= N_PT are padding
    size_t base = (size_t)c * ROW + 64;
    int mb = (lane < 16) ? 0 : 8;
#pragma unroll
    for (int T = 0; T < 8; ++T) {
#pragma unroll
        for (int r2 = 0; r2 < 8; ++r2) {
            int m = mb + r2;
            if (m < N_PT) out[base + (size_t)m * 128 + T * 16 + n] = acc[T][r2];
        }
    }
}

// ---------------------------------------------------------------- launch
extern "C" void kernel_launch(void* const* d_in, const int* in_sizes, int n_in,
                              void* d_out, int out_size, void* d_ws, size_t ws_size,
                              hipStream_t stream) {
    const float* pos   = (const float*)d_in[0];
    const float* reta  = (const float*)d_in[1];
    const float* rmu   = (const float*)d_in[2];
    const float* anu   = (const float*)d_in[3];
    const float* azeta = (const float*)d_in[4];
    const float* aeta  = (const float*)d_in[5];
    const float* amu   = (const float*)d_in[6];
    const int*   elem  = (const int*)d_in[7];
    const int*   pidx  = (const int*)d_in[8];
    const int*   tcen  = (const int*)d_in[9];
    const int*   tpair = (const int*)d_in[10];
    const int*   tflip = (const int*)d_in[11];

    int natoms = in_sizes[0] / 3;
    int npair  = in_sizes[8] / 2;
    int ntri   = in_sizes[9];

    float4* pv = (float4*)d_ws;
    int*    es = (int*)((char*)d_ws + (size_t)npair * sizeof(float4));

    float* out = (float*)d_out;

    // 1) zero output (radial region needs it; angular gets overwritten)
    zero_out_k<<<1024, 256, 0, stream>>>(out, (size_t)out_size);

    // 2) per-pair vectors/distances/species
    if (npair > 0) {
        int blocks = (npair + 255) / 256;
        prep_pairs_k<<<blocks, 256, 0, stream>>>(pos, pidx, elem, pv, es, npair);
    }
    // 3) radial AEV (atomic scatter, small)
    if (npair > 0) {
        int total = npair * 16;
        int blocks = (total + 255) / 256;
        radial_k<<<blocks, 256, 0, stream>>>(pv, es, pidx, reta, rmu, out, npair);
    }
    // 4) angular AEV: one wave per center, one-hot WMMA GEMM accumulation
    angular_k<<<natoms, 32, 0, stream>>>(pv, es, tcen, tpair, tflip,
                                         anu, azeta, aeta, amu, out, ntri);
}